// MorphologicalPrototypePerception_71176198029364
// MI455X (gfx1250) — compile-verified
//
#include <hip/hip_runtime.h>
#include <math.h>

#define Bn 8
#define Cn 64
#define Hn 256
#define Wn 256
#define Ln 65536
#define Pn 16

typedef __attribute__((ext_vector_type(2))) float v2f;
typedef __attribute__((ext_vector_type(8))) float v8f;

// D = A(16x4 f32) * B(4x16 f32) + C(16x16 f32), exact fp32 via CDNA5 WMMA
__device__ __forceinline__ v8f wmma4(v2f a, v2f b, v8f c) {
  return __builtin_amdgcn_wmma_f32_16x16x4_f32(false, a, false, b, (short)0, c,
                                               false, false);
}

// order-preserving float <-> uint for atomicMax
__device__ __forceinline__ unsigned fenc(float f) {
  unsigned u = __float_as_uint(f);
  return (u & 0x80000000u) ? ~u : (u | 0x80000000u);
}
__device__ __forceinline__ float fdec(unsigned u) {
  unsigned b = (u & 0x80000000u) ? (u & 0x7fffffffu) : ~u;
  return __uint_as_float(b);
}

// ---------------- K1: channel-mean map ----------------
__global__ __launch_bounds__(256) void k_meanmap(const float* __restrict__ feat,
                                                 float* __restrict__ meanmap) {
  int b = blockIdx.x >> 8;
  int l = ((blockIdx.x & 255) << 8) | threadIdx.x;
  const float* fp = feat + (size_t)b * Cn * Ln + l;
  float s = 0.f;
#pragma unroll 8
  for (int c = 0; c < Cn; ++c) s += fp[(size_t)c * Ln];
  meanmap[(size_t)b * Ln + l] = s * (1.0f / 64.0f);
}

// ---------------- K2: Sobel + per-batch sum / sumsq ----------------
__global__ __launch_bounds__(256) void k_sobel(const float* __restrict__ meanmap,
                                               float* __restrict__ es,
                                               float* __restrict__ orix,
                                               float* __restrict__ oriy,
                                               float* __restrict__ sums) {
  __shared__ float r1[256], r2[256];
  int tid = threadIdx.x;
  int b = blockIdx.x >> 8;
  int l = ((blockIdx.x & 255) << 8) | tid;
  int y = l >> 8, x = l & 255;
  const float* mp = meanmap + (size_t)b * Ln;
  float v[3][3];
#pragma unroll
  for (int dy = -1; dy <= 1; ++dy)
#pragma unroll
    for (int dx = -1; dx <= 1; ++dx) {
      int yy = y + dy, xx = x + dx;
      v[dy + 1][dx + 1] =
          (yy >= 0 && yy < Hn && xx >= 0 && xx < Wn) ? mp[yy * Wn + xx] : 0.f;
    }
  float gx = -v[0][0] + v[0][2] - 2.f * v[1][0] + 2.f * v[1][2] - v[2][0] + v[2][2];
  float gy = -v[0][0] - 2.f * v[0][1] - v[0][2] + v[2][0] + 2.f * v[2][1] + v[2][2];
  float d = gx * gx + gy * gy + 1e-6f;
  float rd = 1.0f / d;  // one divide instead of two
  float e = sqrtf(d);
  es[(size_t)b * Ln + l] = e;
  orix[(size_t)b * Ln + l] = (gx * gx - gy * gy) * rd;
  oriy[(size_t)b * Ln + l] = 2.f * gx * gy * rd;
  r1[tid] = e;
  r2[tid] = e * e;
  __syncthreads();
  for (int st = 128; st > 0; st >>= 1) {
    if (tid < st) { r1[tid] += r1[tid + st]; r2[tid] += r2[tid + st]; }
    __syncthreads();
  }
  if (tid == 0) {
    atomicAdd(&sums[2 * b], r1[0]);
    atomicAdd(&sums[2 * b + 1], r2[0]);
  }
}

// depthwise 3x3 (SAME, zero pad) + BN1 + ReLU for one (ci, pixel)
__device__ __forceinline__ float dwval(const float* __restrict__ fb,
                                       const float* __restrict__ s_dw,
                                       const float* __restrict__ s_sc1,
                                       const float* __restrict__ s_bi1, int ci,
                                       int y, int x) {
  const float* fp = fb + (size_t)ci * Ln;
  const float* wp = s_dw + ci * 9;
  float s = 0.f;
#pragma unroll
  for (int ky = 0; ky < 3; ++ky) {
    int yy = y + ky - 1;
    if (yy >= 0 && yy < Hn) {
#pragma unroll
      for (int kx = 0; kx < 3; ++kx) {
        int xx = x + kx - 1;
        if (xx >= 0 && xx < Wn) s += wp[ky * 3 + kx] * fp[yy * Wn + xx];
      }
    }
  }
  return fmaxf(s * s_sc1[ci] + s_bi1[ci], 0.f);
}

// ---------------- K3: fused dw->pw->affinity GEMMs + bias + max ----------------
__global__ __launch_bounds__(256) void k_logits(
    const float* __restrict__ feat, const float* __restrict__ dw_w,
    const float* __restrict__ g1, const float* __restrict__ b1,
    const float* __restrict__ m1, const float* __restrict__ v1,
    const float* __restrict__ pw_w, const float* __restrict__ g2,
    const float* __restrict__ b2, const float* __restrict__ m2,
    const float* __restrict__ v2, const float* __restrict__ aff_w,
    const float* __restrict__ tscale_p, const float* __restrict__ oscale_p,
    const float* __restrict__ proto, const float* __restrict__ es,
    const float* __restrict__ orix, const float* __restrict__ oriy,
    const float* __restrict__ sums, float* __restrict__ logits,
    unsigned* __restrict__ maxenc) {
  __shared__ float s_pw[Cn * Cn];
  __shared__ float s_aff[Pn * Cn];
  __shared__ float s_dw[Cn * 9];
  __shared__ float s_sc1[Cn], s_bi1[Cn], s_sc2[Cn], s_bi2[Cn];
  __shared__ float s_pdx[Pn], s_pdy[Pn];
  __shared__ float s_F[8][Pn * Cn];
  __shared__ unsigned s_max[Pn];
  int tid = threadIdx.x;
  int bIdx = blockIdx.y;
  for (int i = tid; i < Cn * Cn; i += 256) s_pw[i] = pw_w[i];
  for (int i = tid; i < Pn * Cn; i += 256) s_aff[i] = aff_w[i];
  for (int i = tid; i < Cn * 9; i += 256) s_dw[i] = dw_w[i];
  if (tid < Cn) {
    float i1 = rsqrtf(v1[tid] + 1e-5f);
    s_sc1[tid] = g1[tid] * i1;
    s_bi1[tid] = b1[tid] - m1[tid] * g1[tid] * i1;
    float i2 = rsqrtf(v2[tid] + 1e-5f);
    s_sc2[tid] = g2[tid] * i2;
    s_bi2[tid] = b2[tid] - m2[tid] * g2[tid] * i2;
  }
  if (tid < Pn) {
    float px = proto[2 * tid], py = proto[2 * tid + 1];
    float nr = fmaxf(sqrtf(px * px + py * py), 1e-12f);
    s_pdx[tid] = px / nr;
    s_pdy[tid] = py / nr;
    s_max[tid] = 0u;
  }
  __syncthreads();
  float se = sums[2 * bIdx], se2 = sums[2 * bIdx + 1];
  float meanE = se * (1.0f / (float)Ln);
  float varE = (se2 - se * se * (1.0f / (float)Ln)) * (1.0f / (float)(Ln - 1));
  float stdE = sqrtf(fmaxf(varE, 0.f));
  // hoisted reciprocals: denominators are uniform per batch, so divide once
  float inv_std = 1.0f / (stdE + 1e-5f);
  float inv_mean = 1.0f / (meanE + 1e-6f);
  float tsc = tscale_p[0], osc = oscale_p[0];
  int wave = tid >> 5, lane = tid & 31;
  int n = lane & 15, kh = lane >> 4;
  const float* fb = feat + (size_t)bIdx * Cn * Ln;
  for (int it = 0; it < 4; ++it) {
    int strip = blockIdx.x * 512 + wave * 64 + it * 16;
    int l = strip + n;
    int y = l >> 8, x = l & 255;
    // GEMM1: X[16pix x 64ic] * pw^T[64ic x 64oc], K in 16 steps of 4
    v8f acc[4] = {{}, {}, {}, {}};
    for (int kk = 0; kk < 16; ++kk) {
      int ci0 = 4 * kk + 2 * kh;  // A: vgpr j -> k = 2*kh + j
      v2f A;
      A.x = dwval(fb, s_dw, s_sc1, s_bi1, ci0, y, x);
      A.y = dwval(fb, s_dw, s_sc1, s_bi1, ci0 + 1, y, x);
#pragma unroll
      for (int t = 0; t < 4; ++t) {
        v2f Bt;  // B[k=2*kh+j][n=oc] = pw_w[oc][ic]
        Bt.x = s_pw[(t * 16 + n) * 64 + 4 * kk + 2 * kh];
        Bt.y = s_pw[(t * 16 + n) * 64 + 4 * kk + 2 * kh + 1];
        acc[t] = wmma4(A, Bt, acc[t]);
      }
    }
    // BN2 + ReLU, transpose through LDS (C-layout: M=pix=r+8*kh, N=oc)
#pragma unroll
    for (int t = 0; t < 4; ++t) {
      int oc = t * 16 + n;
      float sc = s_sc2[oc], bi = s_bi2[oc];
#pragma unroll
      for (int r = 0; r < 8; ++r) {
        int mp = r + 8 * kh;
        s_F[wave][mp * 64 + oc] = fmaxf(acc[t][r] * sc + bi, 0.f);
      }
    }
    __syncthreads();
    // GEMM2: rawaff[16p x 16pix] = aff_w[16p x 64] * F^T[64 x 16pix]
    v8f accA = {};
    for (int kk = 0; kk < 16; ++kk) {
      v2f A2, B2;
      A2.x = s_aff[n * 64 + 4 * kk + 2 * kh];
      A2.y = s_aff[n * 64 + 4 * kk + 2 * kh + 1];
      B2.x = s_F[wave][n * 64 + 4 * kk + 2 * kh];
      B2.y = s_F[wave][n * 64 + 4 * kk + 2 * kh + 1];
      accA = wmma4(A2, B2, accA);
    }
    float ev = es[(size_t)bIdx * Ln + l];
    float ew = (ev - meanE) * inv_std;
    float ec = fminf(fmaxf(ev * inv_mean, 0.f), 3.f);
    float ox = orix[(size_t)bIdx * Ln + l];
    float oy = oriy[(size_t)bIdx * Ln + l];
    float* lp = logits + ((size_t)bIdx * Ln + l) * Pn;
#pragma unroll
    for (int r = 0; r < 8; ++r) {
      int p = r + 8 * kh;
      float lg = accA[r] + tsc * ew + osc * (ox * s_pdx[p] + oy * s_pdy[p]) * ec;
      lp[p] = lg;
      atomicMax(&s_max[p], fenc(lg));
    }
    __syncthreads();
  }
  if (tid < Pn) atomicMax(&maxenc[bIdx * Pn + tid], s_max[tid]);
}

// ---------------- K4: decode per-(b,p) max ----------------
__global__ void k_decode(const unsigned* __restrict__ maxenc,
                         float* __restrict__ maxval) {
  int i = threadIdx.x;
  if (i < Bn * Pn) maxval[i] = fdec(maxenc[i]);
}

// ---------------- K5: softmax numerator + prototype GEMM ----------------
__global__ __launch_bounds__(256) void k_proto(const float* __restrict__ logits,
                                               const float* __restrict__ key,
                                               const float* __restrict__ maxval,
                                               float* __restrict__ sumexp,
                                               float* __restrict__ proto_acc) {
  int tid = threadIdx.x;
  int b = blockIdx.y;
  int wave = tid >> 5, lane = tid & 31;
  int n = lane & 15, kh = lane >> 4;
  float mv = maxval[b * Pn + n];
  const float* lb = logits + (size_t)b * Ln * Pn;
  const float* kb = key + (size_t)b * Ln * Cn;
  v8f acc[4] = {{}, {}, {}, {}};
  float ssum = 0.f;
  int l0w = blockIdx.x * 512 + wave * 64;
  for (int ks = 0; ks < 16; ++ks) {
    int l0 = l0w + ks * 4;
    int la = l0 + 2 * kh;  // both A and B use k = 2*kh + j
    float e0 = __expf(lb[(size_t)la * Pn + n] - mv);
    float e1 = __expf(lb[(size_t)(la + 1) * Pn + n] - mv);
    v2f A;
    A.x = e0;
    A.y = e1;
    ssum += e0 + e1;
#pragma unroll
    for (int t = 0; t < 4; ++t) {
      v2f Bt;
      Bt.x = kb[(size_t)la * Cn + t * 16 + n];
      Bt.y = kb[(size_t)(la + 1) * Cn + t * 16 + n];
      acc[t] = wmma4(A, Bt, acc[t]);
    }
  }
  ssum += __shfl_xor(ssum, 16, 32);
  if (kh == 0) atomicAdd(&sumexp[b * Pn + n], ssum);
#pragma unroll
  for (int t = 0; t < 4; ++t)
#pragma unroll
    for (int r = 0; r < 8; ++r)
      atomicAdd(&proto_acc[((size_t)b * Pn + r + 8 * kh) * Cn + t * 16 + n],
                acc[t][r]);
}

// ---------------- K6: gate MLPs + sigmoid + residual + LayerNorm ----------------
__global__ __launch_bounds__(256) void k_gate(
    const float* __restrict__ proto_acc, const float* __restrict__ sumexp,
    const float* __restrict__ query, const float* __restrict__ lw1,
    const float* __restrict__ lb1, const float* __restrict__ lw2,
    const float* __restrict__ lb2, const float* __restrict__ gw1,
    const float* __restrict__ gb1, const float* __restrict__ gw2,
    const float* __restrict__ gb2, const float* __restrict__ lng,
    const float* __restrict__ lnb, float* __restrict__ out) {
  __shared__ float z[Pn * Cn], h1[Pn * Cn], locv[Pn * Cn];
  __shared__ float gz[Cn], gh[Cn], gv[Cn];
  __shared__ float rse[Pn];
  int tid = threadIdx.x;
  int b = blockIdx.x;
  if (tid < Pn) rse[tid] = 1.0f / sumexp[b * Pn + tid];
  __syncthreads();
  for (int i = tid; i < Pn * Cn; i += 256) {
    int p = i >> 6;
    z[i] = proto_acc[(size_t)b * Pn * Cn + i] * rse[p] +
           query[(size_t)b * Pn * Cn + i];
  }
  __syncthreads();
  for (int i = tid; i < Pn * Cn; i += 256) {
    int p = i >> 6, oc = i & 63;
    float s = lb1[oc];
    for (int ic = 0; ic < Cn; ++ic) s += z[p * 64 + ic] * lw1[oc * 64 + ic];
    h1[i] = fmaxf(s, 0.f);
  }
  __syncthreads();
  for (int i = tid; i < Pn * Cn; i += 256) {
    int p = i >> 6, oc = i & 63;
    float s = lb2[oc];
    for (int ic = 0; ic < Cn; ++ic) s += h1[p * 64 + ic] * lw2[oc * 64 + ic];
    locv[i] = s;
  }
  if (tid < Cn) {
    float s = 0.f;
    for (int p = 0; p < Pn; ++p) s += z[p * 64 + tid];
    gz[tid] = s * (1.0f / 16.0f);
  }
  __syncthreads();
  if (tid < Cn) {
    float s = gb1[tid];
    for (int ic = 0; ic < Cn; ++ic) s += gz[ic] * gw1[tid * 64 + ic];
    gh[tid] = fmaxf(s, 0.f);
  }
  __syncthreads();
  if (tid < Cn) {
    float s = gb2[tid];
    for (int ic = 0; ic < Cn; ++ic) s += gh[ic] * gw2[tid * 64 + ic];
    gv[tid] = s;
  }
  __syncthreads();
  if (tid < Pn) {
    float vals[Cn];
    float mu = 0.f;
    for (int c = 0; c < Cn; ++c) {
      float q = query[(size_t)b * Pn * Cn + tid * 64 + c];
      float aw = 1.0f / (1.0f + __expf(-(locv[tid * 64 + c] + gv[c])));
      float v = q * aw + q;
      vals[c] = v;
      mu += v;
    }
    mu *= (1.0f / 64.0f);
    float var = 0.f;
    for (int c = 0; c < Cn; ++c) {
      float d = vals[c] - mu;
      var += d * d;
    }
    var *= (1.0f / 64.0f);
    float inv = rsqrtf(var + 1e-5f);
    for (int c = 0; c < Cn; ++c)
      out[(size_t)b * Pn * Cn + tid * 64 + c] =
          (vals[c] - mu) * inv * lng[c] + lnb[c];
  }
}

extern "C" void kernel_launch(void* const* d_in, const int* in_sizes, int n_in,
                              void* d_out, int out_size, void* d_ws,
                              size_t ws_size, hipStream_t stream) {
  (void)in_sizes; (void)n_in; (void)out_size; (void)ws_size;
  const float* feat = (const float*)d_in[0];
  const float* query = (const float*)d_in[1];
  const float* key = (const float*)d_in[2];
  const float* dw_w = (const float*)d_in[3];
  const float* g1 = (const float*)d_in[4];
  const float* b1 = (const float*)d_in[5];
  const float* m1 = (const float*)d_in[6];
  const float* v1 = (const float*)d_in[7];
  const float* pw_w = (const float*)d_in[8];
  const float* g2 = (const float*)d_in[9];
  const float* b2 = (const float*)d_in[10];
  const float* m2 = (const float*)d_in[11];
  const float* v2 = (const float*)d_in[12];
  const float* aff_w = (const float*)d_in[13];
  const float* lw1 = (const float*)d_in[14];
  const float* lb1 = (const float*)d_in[15];
  const float* lw2 = (const float*)d_in[16];
  const float* lb2 = (const float*)d_in[17];
  const float* gw1 = (const float*)d_in[18];
  const float* gb1 = (const float*)d_in[19];
  const float* gw2 = (const float*)d_in[20];
  const float* gb2 = (const float*)d_in[21];
  const float* lng = (const float*)d_in[22];
  const float* lnb = (const float*)d_in[23];
  const float* tsc = (const float*)d_in[24];
  const float* osc = (const float*)d_in[25];
  const float* proto = (const float*)d_in[26];

  char* ws = (char*)d_ws;
  float* sums = (float*)(ws + 0);            // 16 floats
  unsigned* maxenc = (unsigned*)(ws + 64);   // 128 u32
  float* sumexp = (float*)(ws + 576);        // 128 floats
  float* maxval = (float*)(ws + 1088);       // 128 floats
  float* proto_acc = (float*)(ws + 1600);    // 8192 floats (ends @34368)
  size_t off = 65536;
  float* meanmap = (float*)(ws + off); off += (size_t)Bn * Ln * 4;
  float* es = (float*)(ws + off);      off += (size_t)Bn * Ln * 4;
  float* orix = (float*)(ws + off);    off += (size_t)Bn * Ln * 4;
  float* oriy = (float*)(ws + off);    off += (size_t)Bn * Ln * 4;
  float* logits = (float*)(ws + off);  off += (size_t)Bn * Ln * Pn * 4;

  hipMemsetAsync(d_ws, 0, 34368, stream);  // zero accumulators each call
  k_meanmap<<<dim3(Bn * 256), 256, 0, stream>>>(feat, meanmap);
  k_sobel<<<dim3(Bn * 256), 256, 0, stream>>>(meanmap, es, orix, oriy, sums);
  k_logits<<<dim3(Ln / 512, Bn), 256, 0, stream>>>(
      feat, dw_w, g1, b1, m1, v1, pw_w, g2, b2, m2, v2, aff_w, tsc, osc, proto,
      es, orix, oriy, sums, logits, maxenc);
  k_decode<<<1, 128, 0, stream>>>(maxenc, maxval);
  k_proto<<<dim3(Ln / 512, Bn), 256, 0, stream>>>(logits, key, maxval, sumexp,
                                                  proto_acc);
  k_gate<<<Bn, 256, 0, stream>>>(proto_acc, sumexp, query, lw1, lb1, lw2, lb2,
                                 gw1, gb1, gw2, gb2, lng, lnb, (float*)d_out);
}